// DiarizationLoss_17386027614816
// MI455X (gfx1250) — compile-verified
//
#include <hip/hip_runtime.h>
#include <hip/hip_bf16.h>

#define B_  32
#define T_  65536
#define S_  4
#define EPS_ 1e-7f
#define LN2_ 0.6931471805599453094f

#define NGRP   8      // batch groups of 4 (block-diagonal WMMA packing)
#define NCHUNK 64     // t-chunks per group
#define NWAVE  8      // waves per block (256 threads)
#define TPW    (T_ / (NCHUNK * NWAVE))   // 128 timesteps per wave
#define NITER  (TPW / 4)                 // 32 WMMA iterations per wave

typedef __attribute__((ext_vector_type(2))) float v2f;
typedef __attribute__((ext_vector_type(8))) float v8f;

// ws layout (floats): [0,512) term1[b][i][j] ; [512,640) term2[b][i] ; [640] vad numerator
#define WS_T1  0
#define WS_T2  512
#define WS_VAD 640

__global__ void zero_ws_kernel(float* __restrict__ ws) {
    for (int i = threadIdx.x; i < 641; i += blockDim.x) ws[i] = 0.0f;
}

// Heavy kernel: per-batch 4x4 cost via V_WMMA_F32_16X16X4_F32, 4 batches packed
// block-diagonally into the 16x16 accumulator, K = 4 consecutive timesteps.
// Logs accumulated in log2 units (native v_log_f32); scaled by ln2 once at the end.
__global__ void pit_cost_kernel(const float* __restrict__ pred,
                                const float* __restrict__ labels,
                                const int*   __restrict__ lengths,
                                float*       __restrict__ ws) {
    __shared__ float sh1[64];   // [bb][i][j]  raw +sum X*(labels*mask)   (ln units)
    __shared__ float sh2[16];   // [bb][i]     raw +sum lq*mask           (ln units)

    const int tid = threadIdx.x;
    if (tid < 64) sh1[tid] = 0.0f;
    if (tid < 16) sh2[tid] = 0.0f;
    __syncthreads();

    const int lane  = tid & 31;
    const int wave  = tid >> 5;
    const int b0    = blockIdx.x * 4;          // batch group base
    const int chunk = blockIdx.y;

    // A-fragment row M == B-fragment col N == lane&15 (shared bb, shared index math)
    const int mn = lane & 15;                  // M = N
    const int bb = mn >> 2;                    // batch within group
    const int ij = mn & 3;                     // i (A side) == j (B side)
    const int kb = (lane < 16) ? 0 : 2;        // K sub-offset per lane half
    const int len = lengths[b0 + bb];

    const int baseRow = ((b0 + bb) * T_) * S_ + ij;

    const int t0   = (chunk * NWAVE + wave) * TPW;
    const int tend = t0 + TPW;

    v8f   c    = {};
    float acc2 = 0.0f;

    if (__all((int)(tend <= len))) {
        // ---- fast path: fully inside every lane's valid range, no masks ----
        int tA = t0 + kb;
#pragma unroll 4
        for (int it = 0; it < NITER; ++it) {
            const int idx = baseRow + tA * S_;
            float p0 = pred[idx];
            float p1 = pred[idx + S_];
            const float l0 = labels[idx];
            const float l1 = labels[idx + S_];
            p0 = fminf(fmaxf(p0, EPS_), 1.0f - EPS_);
            p1 = fminf(fmaxf(p1, EPS_), 1.0f - EPS_);
            const float lp0 = __log2f(p0), lq0 = __log2f(1.0f - p0);
            const float lp1 = __log2f(p1), lq1 = __log2f(1.0f - p1);

            v2f a, bmat;
            a[0]    = lp0 - lq0;  a[1]    = lp1 - lq1;
            bmat[0] = l0;         bmat[1] = l1;
            acc2 += lq0 + lq1;

            c = __builtin_amdgcn_wmma_f32_16x16x4_f32(
                    false, a, false, bmat, (short)0, c, false, false);
            tA += 4;
        }
    } else {
        // ---- masked path ----
        int tA = t0 + kb;
#pragma unroll 2
        for (int it = 0; it < NITER; ++it) {
            const int idx = baseRow + tA * S_;
            float p0 = pred[idx];
            float p1 = pred[idx + S_];
            const float l0 = labels[idx];
            const float l1 = labels[idx + S_];
            const float m0 = (tA     < len) ? 1.0f : 0.0f;
            const float m1 = (tA + 1 < len) ? 1.0f : 0.0f;
            p0 = fminf(fmaxf(p0, EPS_), 1.0f - EPS_);
            p1 = fminf(fmaxf(p1, EPS_), 1.0f - EPS_);
            const float lp0 = __log2f(p0), lq0 = __log2f(1.0f - p0);
            const float lp1 = __log2f(p1), lq1 = __log2f(1.0f - p1);

            v2f a, bmat;
            a[0]    = lp0 - lq0;  a[1]    = lp1 - lq1;
            bmat[0] = l0 * m0;    bmat[1] = l1 * m1;
            acc2 += lq0 * m0 + lq1 * m1;

            c = __builtin_amdgcn_wmma_f32_16x16x4_f32(
                    false, a, false, bmat, (short)0, c, false, false);
            tA += 4;
        }
    }

    // fold this wave into block-level LDS accumulators (convert log2 -> ln here)
    atomicAdd(&sh2[mn], acc2 * LN2_);
#pragma unroll
    for (int r = 0; r < 8; ++r) {
        const int M = r + ((lane < 16) ? 0 : 8);
        const int N = mn;
        if ((M >> 2) == (N >> 2)) {                     // diagonal block only
            atomicAdd(&sh1[(M >> 2) * 16 + (M & 3) * 4 + (N & 3)], c[r] * LN2_);
        }
    }
    __syncthreads();

    // one global atomic per accumulator per block (negate: term1/term2 are -sums)
    if (tid < 64)                 atomicAdd(&ws[WS_T1 + b0 * 16 + tid], -sh1[tid]);
    if (tid >= 64 && tid < 80)    atomicAdd(&ws[WS_T2 + b0 * 4 + (tid - 64)], -sh2[tid - 64]);
}

__global__ void vad_loss_kernel(const float* __restrict__ pv,
                                const float* __restrict__ vd,
                                const int*   __restrict__ lengths,
                                float*       __restrict__ ws) {
    __shared__ int   slen[B_];
    __shared__ float sh[NWAVE];
    if (threadIdx.x < B_) slen[threadIdx.x] = lengths[threadIdx.x];
    __syncthreads();

    const int gid    = blockIdx.x * blockDim.x + threadIdx.x;
    const int stride = gridDim.x * blockDim.x;
    float acc = 0.0f;                         // in log2 units
    for (int i = gid; i < B_ * T_; i += stride) {
        const int b = i >> 16;                // T_ == 65536
        const int t = i & (T_ - 1);
        if (t < slen[b]) {
            float p = fminf(fmaxf(pv[i], EPS_), 1.0f - EPS_);
            const float v  = vd[i];
            const float lp = __log2f(p);
            const float lq = __log2f(1.0f - p);
            acc -= lq + v * (lp - lq);
        }
    }
    for (int off = 16; off > 0; off >>= 1) acc += __shfl_down(acc, off, 32);
    if ((threadIdx.x & 31) == 0) sh[threadIdx.x >> 5] = acc;
    __syncthreads();
    if (threadIdx.x == 0) {
        float s = 0.0f;
        for (int w = 0; w < NWAVE; ++w) s += sh[w];
        atomicAdd(&ws[WS_VAD], s * LN2_);
    }
}

// one wave: lane b handles batch b; 24-permutation search over its 4x4 L matrix
__global__ void finalize_kernel(const int* __restrict__ lengths,
                                const float* __restrict__ ws,
                                float* __restrict__ out) {
    const int lane = threadIdx.x;     // 0..31 == batch id
    const int len  = lengths[lane];
    const float msum = (float)((len < T_) ? len : T_);

    float L[4][4];
#pragma unroll
    for (int i = 0; i < 4; ++i)
#pragma unroll
        for (int j = 0; j < 4; ++j)
            L[i][j] = (ws[WS_T1 + lane * 16 + i * 4 + j] + ws[WS_T2 + lane * 4 + i]) / msum;

    float best = 3.4e38f;
#pragma unroll
    for (int p0 = 0; p0 < 4; ++p0)
#pragma unroll
        for (int p1 = 0; p1 < 4; ++p1) {
            if (p1 == p0) continue;
#pragma unroll
            for (int p2 = 0; p2 < 4; ++p2) {
                if (p2 == p0 || p2 == p1) continue;
                const int p3 = 6 - p0 - p1 - p2;
                const float v = (L[0][p0] + L[1][p1] + L[2][p2] + L[3][p3]) * 0.25f;
                best = fminf(best, v);
            }
        }

    float msums = msum;
    for (int off = 16; off > 0; off >>= 1) {
        best  += __shfl_down(best, off, 32);
        msums += __shfl_down(msums, off, 32);
    }
    if (lane == 0) {
        const float speaker_loss = best / (float)B_;
        const float vad_loss = ws[WS_VAD] / msums;
        out[0] = 1.0f * speaker_loss + 0.5f * vad_loss;
    }
}

extern "C" void kernel_launch(void* const* d_in, const int* in_sizes, int n_in,
                              void* d_out, int out_size, void* d_ws, size_t ws_size,
                              hipStream_t stream) {
    const float* pred_speakers = (const float*)d_in[0];
    const float* pred_vad      = (const float*)d_in[1];
    const float* labels        = (const float*)d_in[2];
    const float* vad           = (const float*)d_in[3];
    const int*   lengths       = (const int*)d_in[4];
    float* ws  = (float*)d_ws;
    float* out = (float*)d_out;

    zero_ws_kernel<<<1, 256, 0, stream>>>(ws);

    dim3 grid(NGRP, NCHUNK);
    pit_cost_kernel<<<grid, 32 * NWAVE, 0, stream>>>(pred_speakers, labels, lengths, ws);

    vad_loss_kernel<<<512, 32 * NWAVE, 0, stream>>>(pred_vad, vad, lengths, ws);

    finalize_kernel<<<1, 32, 0, stream>>>(lengths, ws, out);
}